// RNNModel_35837207118450
// MI455X (gfx1250) — compile-verified
//
#include <hip/hip_runtime.h>

// RNN: h_t = tanh(W_hh h_{t-1} + W_ih x_t + b), out_t = W_out h_t + b_out
// T=4096 serial steps. One wave (32 lanes) owns 16 batch columns.
// Step = 3x V_WMMA_F32_16X16X4_F32 (K=12 padded from 10), tanh on D,
// lane-local D->B repack with 6 ds_swizzle SWAPX16 cross-half moves.

typedef float v2f __attribute__((ext_vector_type(2)));
typedef float v8f __attribute__((ext_vector_type(8)));

#define Tt 4096
#define Bb 2048
#define Hh 10
#define PF 8  // x prefetch depth (timesteps)

__device__ __forceinline__ float fast_tanh(float v) {
#if __has_builtin(__builtin_amdgcn_tanhf)
  return __builtin_amdgcn_tanhf(v);
#else
  // tanh(x) = 1 - 2/(exp2(2*log2e*x)+1); saturates correctly, NaN propagates
  float e = __builtin_amdgcn_exp2f(v * 2.8853900817779268f);
  float r = __builtin_amdgcn_rcpf(e + 1.0f);
  return __builtin_fmaf(-2.0f, r, 1.0f);
#endif
}

__device__ __forceinline__ float swapx16(float v) {
  // ds_swizzle group-of-32: xor=0x10, or=0, and=0x1f -> swap half-waves
  return __int_as_float(__builtin_amdgcn_ds_swizzle(__float_as_int(v), 0x401f));
}

__global__ void __launch_bounds__(32) rnn1250_kernel(
    const float* __restrict__ x,    // [T,B]
    const float* __restrict__ h0,   // [B,H]
    const float* __restrict__ Wih,  // [H]
    const float* __restrict__ bih,  // [H]
    const float* __restrict__ Whh,  // [H,H] row-major, z_j = sum_k W[j,k] h_k
    const float* __restrict__ bhh,  // [H]
    const float* __restrict__ Wout, // [H]
    const float* __restrict__ bout, // [1]
    float* __restrict__ out,        // [T,B]
    float* __restrict__ hlast)      // [B,H]
{
  const int lane = threadIdx.x & 31;
  const int half = lane >> 4;     // 0 = low half-wave, 1 = high
  const int n16  = lane & 15;     // batch column within tile
  const bool lo  = (half == 0);
  const int base = blockIdx.x * 16;
  const int n    = base + n16;

  // ---- A operands: W_hh split into 3 K-chunks of 4, 16x4 f32 WMMA layout.
  // lane M = n16 (both halves); VGPR0 = K=4c+2*half, VGPR1 = K=4c+1+2*half.
  v2f A0, A1, A2;
  {
    const int M = n16;
    auto w = [&](int k) -> float {
      return (M < Hh && k < Hh) ? Whh[M * Hh + k] : 0.0f;
    };
    A0.x = w(0 + 2 * half); A0.y = w(1 + 2 * half);
    A1.x = w(4 + 2 * half); A1.y = w(5 + 2 * half);
    A2.x = w(8 + 2 * half); A2.y = w(9 + 2 * half);
  }

  // ---- per-lane C/D-layout constants: row m = v + 8*half
  float biasv[8], wihv[8], woutv[8];
#pragma unroll
  for (int v = 0; v < 8; ++v) {
    int m = v + 8 * half;
    bool ok = (m < Hh);
    biasv[v] = ok ? (bih[m] + bhh[m]) : 0.0f;
    wihv[v]  = ok ? Wih[m] : 0.0f;
    woutv[v] = ok ? Wout[m] : 0.0f;
  }
  const float w8 = Wout[8], w9 = Wout[9], bo = bout[0];

  // ---- initial hidden state into C/D layout
  v8f H;
#pragma unroll
  for (int v = 0; v < 8; ++v) {
    int m = v + 8 * half;
    H[v] = (m < Hh) ? h0[n * Hh + m] : 0.0f;
  }

  // ---- B operands (4x16 f32 layout) + cross-half copies of h8,h9 (on lo lanes)
  v2f B0, B1, B2;
  float s0 = 0.0f, s1 = 0.0f;
  auto formB = [&]() {
    s0 = swapx16(H[0]);          // lo lanes receive h8
    s1 = swapx16(H[1]);          // lo lanes receive h9
    float s2 = swapx16(H[2]);    // hi lanes receive h2
    float s3 = swapx16(H[3]);    // hi lanes receive h3
    float s6 = swapx16(H[6]);    // hi lanes receive h6
    float s7 = swapx16(H[7]);    // hi lanes receive h7
    B0.x = lo ? H[0] : s2;  B0.y = lo ? H[1] : s3;   // K=0,1 | K=2,3
    B1.x = lo ? H[4] : s6;  B1.y = lo ? H[5] : s7;   // K=4,5 | K=6,7
    B2.x = lo ? s0 : 0.0f;  B2.y = lo ? s1 : 0.0f;   // K=8,9 | K=10,11(=0)
  };
  formB();

  const float* xp = x + base + n16;   // x[t*B + n]; both halves read same addr
  float* op = out + base + n16;

  // ---- software-pipelined x prefetch, PF timesteps deep
  float xc[PF];
#pragma unroll
  for (int i = 0; i < PF; ++i) { xc[i] = *xp; xp += Bb; }

  for (int tb = 0; tb < Tt; tb += PF) {
    float xn[PF];
    if (tb + PF < Tt) {
#pragma unroll
      for (int i = 0; i < PF; ++i) { xn[i] = *xp; xp += Bb; }
    } else {
#pragma unroll
      for (int i = 0; i < PF; ++i) xn[i] = 0.0f;
    }
#pragma unroll
    for (int i = 0; i < PF; ++i) {
      const float xv = xc[i];
      // C = b + W_ih * x_t (per-row, broadcast along batch)
      v8f C;
#pragma unroll
      for (int v = 0; v < 8; ++v) C[v] = __builtin_fmaf(wihv[v], xv, biasv[v]);
      // D = W_hh * H + C via 3 chained K=4 f32 WMMAs
      v8f D = __builtin_amdgcn_wmma_f32_16x16x4_f32(false, A0, false, B0,
                                                    (short)0, C, false, false);
      D = __builtin_amdgcn_wmma_f32_16x16x4_f32(false, A1, false, B1,
                                                (short)0, D, false, false);
      D = __builtin_amdgcn_wmma_f32_16x16x4_f32(false, A2, false, B2,
                                                (short)0, D, false, false);
#pragma unroll
      for (int v = 0; v < 8; ++v) H[v] = fast_tanh(D[v]);
      formB();  // also refreshes s0,s1 = h8,h9 on lo lanes
      // output projection: lo lanes hold full h (h0..7 local + h8,h9 swizzled)
      float o = bo;
#pragma unroll
      for (int v = 0; v < 8; ++v) o = __builtin_fmaf(woutv[v], H[v], o);
      o = __builtin_fmaf(w8, s0, o);
      o = __builtin_fmaf(w9, s1, o);
      if (lo) *op = o;
      op += Bb;
    }
#pragma unroll
    for (int i = 0; i < PF; ++i) xc[i] = xn[i];
  }

  // ---- h_last: lo lanes hold the complete final hidden vector
  if (lo) {
    float* hp = hlast + n * Hh;
#pragma unroll
    for (int v = 0; v < 8; ++v) hp[v] = H[v];
    hp[8] = s0;
    hp[9] = s1;
  }
}

extern "C" void kernel_launch(void* const* d_in, const int* in_sizes, int n_in,
                              void* d_out, int out_size, void* d_ws, size_t ws_size,
                              hipStream_t stream) {
  (void)in_sizes; (void)n_in; (void)d_ws; (void)ws_size; (void)out_size;
  const float* x    = (const float*)d_in[0];
  const float* h0   = (const float*)d_in[1];
  const float* Wih  = (const float*)d_in[2];
  const float* bih  = (const float*)d_in[3];
  const float* Whh  = (const float*)d_in[4];
  const float* bhh  = (const float*)d_in[5];
  const float* Wout = (const float*)d_in[6];
  const float* bout = (const float*)d_in[7];
  float* out   = (float*)d_out;                 // [T*B] flat, then h_last
  float* hlast = out + (size_t)Tt * (size_t)Bb; // [B*H]
  rnn1250_kernel<<<Bb / 16, 32, 0, stream>>>(x, h0, Wih, bih, Whh, bhh,
                                             Wout, bout, out, hlast);
}